// EAMPotential_1692217114988
// MI455X (gfx1250) — compile-verified
//
#include <hip/hip_runtime.h>
#include <stdint.h>

// EAM potential, CDNA5 (gfx1250, wave32).
// B=32 structures, N=1024 atoms, M=512 neighbors, 3 pair types, 2 atom types.
// HBM-bound target: 8 B/element (f32 distance + i32 pair_type), 16.7M elements
// => ~5.5us floor at 23.3 TB/s (fits 192MB L2 across replays).
// Per element after folding: 1 shift + 1 ds_load_b128 (param row) + 2 fma +
// 2 v_exp_f32 + 2 add  ==> well under the memory roofline.

#define NSTRUCT 32
#define NATOM   1024
#define NNEIGH  512
#define R0_INV  (1.0f / 2.7f)
#define LOG2E   1.4426950408889634f

__global__ __launch_bounds__(256) void eam_atoms_kernel(
    const float* __restrict__ dist,   // [B*N*M]
    const float* __restrict__ A,      // [3]
    const float* __restrict__ p,      // [3]
    const float* __restrict__ xi,     // [3]
    const float* __restrict__ q,      // [3]
    const float* __restrict__ s,      // [2]
    const int*   __restrict__ types,  // [B*N]
    const int*   __restrict__ ptypes, // [B*N*M]
    float*       __restrict__ partials) // [gridDim.x] per-block energy sums
{
    // Per-wave staging: 4 chunks x (512B dist + 512B ptype) = 4KB/wave, 32KB/block.
    __shared__ __align__(16) unsigned char smem[8 * 4 * 1024];
    // Pair-type table: 3 rows x {k1, b1, k2, b2}; rows hit disjoint LDS banks.
    __shared__ __align__(16) float ptab[12];
    __shared__ float s_sh[2];
    __shared__ float wave_e[8];

    const int tid  = threadIdx.x;
    const int lane = tid & 31;
    const int wave = tid >> 5;

    const int atom = blockIdx.x * 8 + wave;            // [0, B*N)
    const uint64_t gbyte = (uint64_t)atom * (NNEIGH * 4u);
    const uint64_t gd = (uint64_t)(uintptr_t)dist   + gbyte + (uint64_t)lane * 16u;
    const uint64_t gp = (uint64_t)(uintptr_t)ptypes + gbyte + (uint64_t)lane * 16u;

    // Low 32 bits of a generic pointer into LDS == wave-relative LDS byte address.
    const unsigned lds_lane = (unsigned)(uintptr_t)smem + (unsigned)wave * 4096u
                            + (unsigned)lane * 16u;

    // ---- Issue ALL async global->LDS transfers up front (8 x b128 per lane,
    // 4KB per wave in flight, tracked by ASYNCcnt). ----
#pragma unroll
    for (int c = 0; c < 4; ++c) {
        unsigned ld = lds_lane + (unsigned)c * 1024u;        // distances chunk c
        unsigned lp = ld + 512u;                             // pair_types chunk c
        uint64_t ad = gd + (uint64_t)c * 512u;
        uint64_t ap = gp + (uint64_t)c * 512u;
        asm volatile("global_load_async_to_lds_b128 %0, %1, off"
                     :: "v"(ld), "v"(ad) : "memory");
        asm volatile("global_load_async_to_lds_b128 %0, %1, off"
                     :: "v"(lp), "v"(ap) : "memory");
    }

    // ---- Build the folded pair-type table (3 threads) and stage s[] ----
    // phi = A*exp(-p*x), x = d/R0 - 1  ==>  phi = exp2(k1*d + b1) with
    //   c1 = -p*log2e,  k1 = c1/R0,  b1 = log2(A) - c1
    // rho = xi^2*exp(-2q*x)           ==>  rho = exp2(k2*d + b2) with
    //   c2 = -2q*log2e, k2 = c2/R0,  b2 = 2*log2(xi) - c2
    if (tid < 3) {
        const float c1 = -p[tid] * LOG2E;
        const float c2 = -2.0f * q[tid] * LOG2E;
        ptab[tid * 4 + 0] = c1 * R0_INV;
        ptab[tid * 4 + 1] = __builtin_amdgcn_logf(A[tid]) - c1;
        ptab[tid * 4 + 2] = c2 * R0_INV;
        ptab[tid * 4 + 3] = 2.0f * __builtin_amdgcn_logf(xi[tid]) - c2;
    }
    if (tid < 2) s_sh[tid] = s[tid];
    __syncthreads();

    const char* ptab_b = (const char*)ptab;
    float phi = 0.0f, rho = 0.0f;

    // ---- Drain chunks as they land: wait asynccnt <= 6,4,2,0 ----
#pragma unroll
    for (int c = 0; c < 4; ++c) {
        asm volatile("s_wait_asynccnt %0" :: "n"(6 - 2 * c) : "memory");

        const float4* sd = (const float4*)(smem + wave * 4096 + c * 1024 + lane * 16);
        const int4*   sp = (const int4*)  (smem + wave * 4096 + c * 1024 + 512 + lane * 16);
        const float4 dv = *sd;   // ds_load_b128
        const int4   pv = *sp;   // ds_load_b128

        const float dvals[4] = {dv.x, dv.y, dv.z, dv.w};
        const int   tvals[4] = {pv.x, pv.y, pv.z, pv.w};
#pragma unroll
        for (int j = 0; j < 4; ++j) {
            // One ds_load_b128 fetches the whole folded param row for this pair.
            const float4 t = *(const float4*)(ptab_b + (tvals[j] << 4));
            const float d  = dvals[j];
            phi += __builtin_amdgcn_exp2f(__builtin_fmaf(t.x, d, t.y)); // v_exp_f32
            rho += __builtin_amdgcn_exp2f(__builtin_fmaf(t.z, d, t.w)); // v_exp_f32
        }
    }

    // ---- Deterministic wave32 butterfly reduction ----
#pragma unroll
    for (int off = 16; off >= 1; off >>= 1) {
        phi += __shfl_xor(phi, off, 32);
        rho += __shfl_xor(rho, off, 32);
    }

    if (lane == 0) {
        const int t = types[atom];
        const float sv = (t == 0) ? s_sh[0] : s_sh[1];
        wave_e[wave] = phi - sv * __builtin_sqrtf(rho);
    }
    __syncthreads();

    if (tid == 0) {
        float e = 0.0f;
#pragma unroll
        for (int w = 0; w < 8; ++w) e += wave_e[w];   // fixed order: deterministic
        partials[blockIdx.x] = e;
    }
}

// All 8 atoms of a block share one structure (1024 % 8 == 0), so partials
// [b*128 .. b*128+127] belong to structure b. Fixed-tree reduce + scale by 1/N.
__global__ __launch_bounds__(128) void eam_reduce_kernel(
    const float* __restrict__ partials, float* __restrict__ out)
{
    __shared__ float sdata[128];
    const int t = threadIdx.x;
    sdata[t] = partials[blockIdx.x * 128 + t];
    __syncthreads();
#pragma unroll
    for (int sft = 64; sft > 0; sft >>= 1) {
        if (t < sft) sdata[t] += sdata[t + sft];
        __syncthreads();
    }
    if (t == 0) out[blockIdx.x] = sdata[0] * (1.0f / (float)NATOM);
}

extern "C" void kernel_launch(void* const* d_in, const int* in_sizes, int n_in,
                              void* d_out, int out_size, void* d_ws, size_t ws_size,
                              hipStream_t stream) {
    const float* dist   = (const float*)d_in[0];
    const float* A      = (const float*)d_in[1];
    const float* p      = (const float*)d_in[2];
    const float* xi     = (const float*)d_in[3];
    const float* q      = (const float*)d_in[4];
    const float* s      = (const float*)d_in[5];
    const int*   types  = (const int*)d_in[6];
    const int*   ptypes = (const int*)d_in[7];
    float* out = (float*)d_out;
    float* partials = (float*)d_ws;  // 4096 floats = 16KB scratch

    const int nblocks = (NSTRUCT * NATOM) / 8;   // 4096 blocks, 8 waves each
    eam_atoms_kernel<<<nblocks, 256, 0, stream>>>(dist, A, p, xi, q, s, types,
                                                  ptypes, partials);
    eam_reduce_kernel<<<NSTRUCT, 128, 0, stream>>>(partials, out);
}